// fusionNet_47004122087994
// MI455X (gfx1250) — compile-verified
//
#include <hip/hip_runtime.h>

typedef __attribute__((ext_vector_type(2))) float v2f;
typedef __attribute__((ext_vector_type(8))) float v8f;

#define NB   32      // batch
#define CIN  640     // input channels (K)
#define COUT 5       // output channels (M, padded to 16 in the WMMA tile)
#define TT   1450    // time steps (N)
#define JT   4                              // T tiles per wave (16 cols each)
#define TTILES ((TT + 15) / 16)             // 91
#define TGROUPS ((TTILES + JT - 1) / JT)    // 23

// ---------------------------------------------------------------------------
// Kernel 1: z[b,o,t] = 64 * sum_c W[o,c] * spike[b,c,t]
// One wave32 per (batch, 64-wide T group): 4 accumulators share one A frag.
// No masking in the hot loop: A rows >= COUT and B cols >= TT are index-
// clamped; the garbage lands in D rows/cols that are never stored (WMMA dot
// products are independent per (m,n)). EXEC stays all-ones throughout.
// ---------------------------------------------------------------------------
__global__ __launch_bounds__(32)
void lif_gemm_wmma(const float* __restrict__ spike,   // [NB, CIN, TT]
                   const float* __restrict__ W,       // [COUT, CIN]
                   float* __restrict__ z)             // [NB, COUT, TT]
{
    const int grp  = blockIdx.x;            // 0..TGROUPS-1
    const int b    = blockIdx.y;            // 0..NB-1
    const int lane = threadIdx.x;           // 0..31
    const int half = lane >> 4;             // 0: lanes 0-15, 1: lanes 16-31
    const int lid  = lane & 15;             // row (A) / column (B) index

    // A row, clamped (rows 5..15 duplicate row 4; results discarded at store)
    const int    mrow = (lid < COUT) ? lid : (COUT - 1);
    const float* wrow = W + (size_t)mrow * CIN + 2 * half;  // K pair base

    // 4 T columns for this lane, clamped for loads; validity kept for stores
    const int t0 = grp * (16 * JT) + lid;
    int  tc[JT];
    bool tv[JT];
    #pragma unroll
    for (int j = 0; j < JT; ++j) {
        const int t = t0 + 16 * j;
        tv[j] = (t < TT);
        tc[j] = tv[j] ? t : (TT - 1);
    }

    const float* spike_b = spike + (size_t)b * CIN * TT;

    v8f acc[JT];
    #pragma unroll
    for (int j = 0; j < JT; ++j) acc[j] = (v8f){};

    for (int k = 0; k < CIN; k += 4) {
        // A fragment: 16x4 f32. VGPR0 -> K=k+2*half, VGPR1 -> K=k+2*half+1
        // (contiguous pair -> single global_load_b64; W is L2-resident)
        v2f a;
        a.x = wrow[k];
        a.y = wrow[k + 1];

        // B fragments: 4x16 f32 per tile. Lane = N, VGPR j -> K row.
        // Spike is streamed exactly once -> non-temporal loads.
        const float* s0 = spike_b + (size_t)(k + 2 * half) * TT;
        const float* s1 = s0 + TT;

        #pragma unroll
        for (int j = 0; j < JT; ++j) {
            v2f bb;
            bb.x = __builtin_nontemporal_load(s0 + tc[j]);
            bb.y = __builtin_nontemporal_load(s1 + tc[j]);
            acc[j] = __builtin_amdgcn_wmma_f32_16x16x4_f32(
                         /*neg_a=*/false, a, /*neg_b=*/false, bb,
                         /*c_mod=*/(short)0, acc[j],
                         /*reuse_a=*/false, /*reuse_b=*/false);
        }
    }

    // Store rows 0..COUT-1: C/D layout -> VGPR r = row r, lanes 0-15 = N.
    if (half == 0) {
        float* zb = z + (size_t)b * COUT * TT;
        #pragma unroll
        for (int j = 0; j < JT; ++j) {
            if (tv[j]) {
                #pragma unroll
                for (int r = 0; r < COUT; ++r) {
                    zb[(size_t)r * TT + tc[j]] = 64.0f * acc[j][r];  // W_SCALE
                }
            }
        }
    }
}

// ---------------------------------------------------------------------------
// Kernel 2: CUBA-LIF scan over T, fused with delayShift(1).
// One thread per (b,o) channel; 160 serial recurrences of length TT.
// Latency-bound on the v-chain (fma -> cmp -> select); kept minimal.
// ---------------------------------------------------------------------------
__global__ __launch_bounds__(64)
void lif_scan(const float* __restrict__ z,   // [NB, COUT, TT]
              float* __restrict__ out)       // [NB, COUT, TT]
{
    const int idx = blockIdx.x * blockDim.x + threadIdx.x;
    if (idx >= NB * COUT) return;

    const float a_i   = (4096.0f - 1024.0f) / 4096.0f;  // 0.75
    const float a_v   = (4096.0f - 128.0f)  / 4096.0f;  // 0.96875
    const float theta = 80.0f * 64.0f;                  // 5120

    const float* zc = z   + (size_t)idx * TT;
    float*       oc = out + (size_t)idx * TT;

    float u = 0.0f, v = 0.0f;
    float prev_s = 0.0f;                // delayShift: out[t] = s[t-1], out[0]=0

    for (int t = 0; t < TT; ++t) {
        oc[t] = prev_s;
        u = __builtin_fmaf(u, a_i, zc[t]);          // chain: 1 fma (parallel)
        float vn = __builtin_fmaf(v, a_v, u);       // chain: fma
        const bool sp = (vn >= theta);              // chain: cmp
        v = sp ? 0.0f : vn;                         // chain: cndmask (reset)
        prev_s = sp ? 1.0f : 0.0f;                  // off-chain
    }
}

// ---------------------------------------------------------------------------
extern "C" void kernel_launch(void* const* d_in, const int* in_sizes, int n_in,
                              void* d_out, int out_size, void* d_ws, size_t ws_size,
                              hipStream_t stream) {
    const float* spike = (const float*)d_in[0];   // [32, 640, 1450]
    const float* W     = (const float*)d_in[1];   // [5, 640]
    float*       out   = (float*)d_out;           // [32, 5, 1450]
    float*       z     = (float*)d_ws;            // [32, 5, 1450] scratch (928 KB)

    dim3 grid(TGROUPS, NB);
    lif_gemm_wmma<<<grid, 32, 0, stream>>>(spike, W, z);

    const int nchan = NB * COUT;                  // 160
    lif_scan<<<(nchan + 63) / 64, 64, 0, stream>>>(z, out);
}